// GATLayer_26637387170384
// MI455X (gfx1250) — compile-verified
//
#include <hip/hip_runtime.h>
#include <math.h>

typedef __attribute__((ext_vector_type(2))) float v2f;
typedef __attribute__((ext_vector_type(8))) float v8f;

#define HH 4
#define DD 32
#define NN 1024
#define BB 8
#define KDIM 128
#define NEG_SLOPE 0.2f

// -------------------- Kernel 1: h = x @ W^T  (8192x128 = 8192x128 * 128x128)
// Block: 256 thr (8 waves). Block -> 16 rows; wave w -> 16 cols (w*16).
// WMMA f32 16x16x4, K chunked by 32 through LDS.
__global__ __launch_bounds__(256) void gat_h_gemm(const float* __restrict__ x,
                                                  const float* __restrict__ W,
                                                  float* __restrict__ h) {
    __shared__ float lds_x[16][132];   // padded: 132 % 64 = 4 -> conflict-free row reads
    __shared__ float lds_w[128][36];   // 36 % 64 -> distinct banks for 16 rows
    const int tid  = threadIdx.x;
    const int lane = tid & 31;
    const int wave = tid >> 5;
    const int r0   = blockIdx.x * 16;
    const int c0   = wave * 16;
    const int lm   = lane & 15;   // M for A-frag / N for B-frag
    const int lhi  = lane >> 4;   // K-pair select

    for (int idx = tid; idx < 16 * KDIM; idx += 256) {
        int row = idx >> 7, col = idx & 127;
        lds_x[row][col] = x[(size_t)(r0 + row) * KDIM + col];
    }

    v8f acc = {};
    for (int kk = 0; kk < KDIM; kk += 32) {
        __syncthreads();
        for (int idx = tid; idx < 128 * 32; idx += 256) {
            int row = idx >> 5, col = idx & 31;
            lds_w[row][col] = W[(size_t)row * KDIM + kk + col];
        }
        if (kk + 32 < KDIM)   // global_prefetch_b8 of next K chunk
            __builtin_prefetch(&W[(size_t)(tid & 127) * KDIM + kk + 32], 0, 1);
        __syncthreads();
#pragma unroll
        for (int s = 0; s < 8; ++s) {
            const int kb = 4 * s + lhi * 2;
            v2f a, b;
            a.x = lds_x[lm][kk + kb + 0];
            a.y = lds_x[lm][kk + kb + 1];
            b.x = lds_w[c0 + lm][kb + 0];
            b.y = lds_w[c0 + lm][kb + 1];
            acc = __builtin_amdgcn_wmma_f32_16x16x4_f32(false, a, false, b,
                                                        (short)0, acc, false, false);
        }
    }
#pragma unroll
    for (int v = 0; v < 8; ++v) {
        int m = v + lhi * 8;                      // D layout: vgpr v, lane-hi -> M
        h[(size_t)(r0 + m) * KDIM + c0 + lm] = acc[v];
    }
}

// -------------------- Kernel 2: alpha_src/dst[bn][h] = h[bn, h*32:+32] . a_{src,dst}[h]
__global__ __launch_bounds__(256) void gat_alpha(const float* __restrict__ h,
                                                 const float* __restrict__ a_src,
                                                 const float* __restrict__ a_dst,
                                                 float* __restrict__ alpha_src,
                                                 float* __restrict__ alpha_dst) {
    int t = blockIdx.x * 256 + threadIdx.x;      // t = bn*4 + head, t < 32768
    int head = t & 3;
    int bn   = t >> 2;
    const float4* hp = (const float4*)(h + (size_t)bn * KDIM + head * DD);
    const float4* sp = (const float4*)(a_src + head * DD);
    const float4* dp = (const float4*)(a_dst + head * DD);
    float ss = 0.f, sd = 0.f;
#pragma unroll
    for (int i = 0; i < 8; ++i) {
        float4 hv = hp[i], av = sp[i], dv = dp[i];
        ss += hv.x * av.x + hv.y * av.y + hv.z * av.z + hv.w * av.w;
        sd += hv.x * dv.x + hv.y * dv.y + hv.z * dv.z + hv.w * dv.w;
    }
    alpha_src[t] = ss;
    alpha_dst[t] = sd;
}

// -------------------- Kernel 3: Dmax[b,h] = max over alive j of alpha_dst[b,j,h]
__global__ __launch_bounds__(256) void gat_dmax(const float* __restrict__ alpha_dst,
                                                const float* __restrict__ alive,
                                                float* __restrict__ dmax) {
    __shared__ float red[256];
    const int bh = blockIdx.x;              // b*4 + head
    const int b = bh >> 2, head = bh & 3;
    float m = -INFINITY;
    for (int j = threadIdx.x; j < NN; j += 256) {
        float d  = alpha_dst[(size_t)(b * NN + j) * HH + head];
        if (alive[b * NN + j] >= 0.5f && d > m) m = d;
    }
    red[threadIdx.x] = m;
    __syncthreads();
    for (int s = 128; s > 0; s >>= 1) {
        if (threadIdx.x < s) red[threadIdx.x] = fmaxf(red[threadIdx.x], red[threadIdx.x + s]);
        __syncthreads();
    }
    if (threadIdx.x == 0) dmax[bh] = red[0];
}

// -------------------- Kernel 4: fused attention. Block -> (b, 16 queries).
// Wave w -> head = w&3, d-half = w>>2. One-pass exp + WMMA P@V, Z folded at store.
__global__ __launch_bounds__(256) void gat_attn(const float* __restrict__ h,
                                                const float* __restrict__ alpha_src,
                                                const float* __restrict__ alpha_dst,
                                                const float* __restrict__ alive,
                                                const float* __restrict__ dmax,
                                                float* __restrict__ out) {
    __shared__ float dval[HH][NN];          // masked d: dead key -> -inf -> exp -> 0
    __shared__ float inv_s[8][16];
    const int tid   = threadIdx.x;
    const int lane  = tid & 31;
    const int wave  = tid >> 5;
    const int head  = wave & 3;
    const int dhalf = wave >> 2;
    const int b     = blockIdx.x >> 6;
    const int q0    = (blockIdx.x & 63) * 16;
    const int lm    = lane & 15;
    const int lhi   = lane >> 4;
    const int d0    = head * DD + dhalf * 16;

    for (int idx = tid; idx < HH * NN; idx += 256) {
        int hh = idx & 3, j = idx >> 2;
        float d = alpha_dst[(size_t)(b * NN + j) * HH + hh];
        dval[hh][j] = (alive[b * NN + j] >= 0.5f) ? d : -INFINITY;
    }
    __syncthreads();

    const float Dm  = dmax[b * HH + head];
    const float s_i = alpha_src[(size_t)(b * NN + q0 + lm) * HH + head];
    const float e0  = s_i + Dm;
    const float m_i = (e0 >= 0.f) ? e0 : NEG_SLOPE * e0;   // exact row max (leaky monotone)

    v8f acc = {};
    float zacc = 0.f;
    if (Dm > -INFINITY) {                    // wave-uniform: EXEC stays all-ones for WMMA
        const float* hbase = h + (size_t)b * NN * KDIM + d0 + lm;
        for (int j0 = 0; j0 < NN; j0 += 4) {
            const int ja = j0 + lhi * 2;
            float ea = s_i + dval[head][ja];
            float eb = s_i + dval[head][ja + 1];
            ea = (ea >= 0.f) ? ea : NEG_SLOPE * ea;
            eb = (eb >= 0.f) ? eb : NEG_SLOPE * eb;
            v2f p, vv;
            p.x = __expf(ea - m_i);
            p.y = __expf(eb - m_i);
            zacc += p.x + p.y;
            vv.x = hbase[(size_t)ja * KDIM];
            vv.y = hbase[(size_t)(ja + 1) * KDIM];
            acc = __builtin_amdgcn_wmma_f32_16x16x4_f32(false, p, false, vv,
                                                        (short)0, acc, false, false);
        }
    }
    zacc += __shfl_xor(zacc, 16, 32);                       // fold K halves -> Z per row
    if (lhi == 0) inv_s[wave][lm] = (zacc > 0.f) ? (1.f / zacc) : 0.f;  // 0 matches nan_to_num
    __syncthreads();
#pragma unroll
    for (int v = 0; v < 8; ++v) {
        int m = v + lhi * 8;
        out[((size_t)(b * NN) + q0 + m) * KDIM + d0 + lm] = acc[v] * inv_s[wave][m];
    }
}

extern "C" void kernel_launch(void* const* d_in, const int* in_sizes, int n_in,
                              void* d_out, int out_size, void* d_ws, size_t ws_size,
                              hipStream_t stream) {
    const float* x     = (const float*)d_in[0];   // [8,1024,128]
    const float* alive = (const float*)d_in[1];   // [8,1024]
    const float* W     = (const float*)d_in[2];   // [128,128]
    const float* a_src = (const float*)d_in[3];   // [4,32]
    const float* a_dst = (const float*)d_in[4];   // [4,32]
    float* out = (float*)d_out;                   // [8,1024,128]

    float* h_ws  = (float*)d_ws;                  // 8192*128 floats = 4 MB
    float* as_ws = h_ws + (size_t)BB * NN * KDIM; // 32768 floats
    float* ad_ws = as_ws + (size_t)BB * NN * HH;  // 32768 floats
    float* dm_ws = ad_ws + (size_t)BB * NN * HH;  // 32 floats

    gat_h_gemm<<<512, 256, 0, stream>>>(x, W, h_ws);
    gat_alpha <<<128, 256, 0, stream>>>(h_ws, a_src, a_dst, as_ws, ad_ws);
    gat_dmax  <<<32,  256, 0, stream>>>(ad_ws, alive, dm_ws);
    gat_attn  <<<512, 256, 0, stream>>>(h_ws, as_ws, ad_ws, alive, dm_ws, out);
}